// CenterExtractor_11433202942250
// MI455X (gfx1250) — compile-verified
//
#include <hip/hip_runtime.h>

// CenterExtractor for MI455X (gfx1250, wave32).
//
// Segmented argmax per (batch, zone): pack (monotonic-float<<32)|~index into
// a u64 so a single 64-bit atomicMax yields max-value with smallest-index
// tie-break (identical to segment_max + segment_min in the reference).
// Pass A pre-reduces into a 64KB LDS zone table (ds_max_u64) before flushing
// to the global 16x8192 table (L2-resident, 1MB). Pass B recomputes the pack
// per pixel and compares against the winner; count via wave32 ballot.
// All streaming traffic uses b128 vector accesses (4 pixels/thread) to
// saturate the 23.3 TB/s HBM path with minimal issue overhead.
//
// No WMMA: this op has no matmul structure; the CDNA5-specific paths used are
// ds_*_u64 LDS atomics, wave32 ballot, global_prefetch_b8, b128 vmem.

#define NUM_ZONES   8192
#define NBATCH      16
#define NPIX        (1024 * 1024)
#define WGS_PER_IMG 32
#define PIX_PER_WG  (NPIX / WGS_PER_IMG)         // 32768
#define THREADS     256
#define VEC         4
#define ITERS       (PIX_PER_WG / (THREADS*VEC)) // 32

__device__ __forceinline__ unsigned long long pack_vi(float v, unsigned p) {
    // Monotonic float->uint mapping (preserves total order for non-NaN):
    unsigned bits = __float_as_uint(v);
    unsigned ord  = (bits & 0x80000000u) ? ~bits : (bits | 0x80000000u);
    // Low 32 bits: ~p, so equal values tie-break to the SMALLEST index under max().
    return ((unsigned long long)ord << 32) | (unsigned long long)(~p);
}

__global__ void init_ws_kernel(unsigned long long* __restrict__ zmax,
                               unsigned* __restrict__ counter) {
    // 16*8192 u64 entries, 4 per thread -> 32768 threads.
    unsigned i = (blockIdx.x * blockDim.x + threadIdx.x) * 4u;
    if (i < NBATCH * NUM_ZONES) {
        zmax[i + 0] = 0ull;   // 0 == "empty" sentinel
        zmax[i + 1] = 0ull;
        zmax[i + 2] = 0ull;
        zmax[i + 3] = 0ull;
    }
    if (i == 0) *counter = 0u;
}

__global__ __launch_bounds__(THREADS) void zone_max_kernel(
        const float* __restrict__ x,
        const int*   __restrict__ zones,
        unsigned long long* __restrict__ zmax) {
    __shared__ unsigned long long lmax[NUM_ZONES];   // 64 KB LDS zone table
    const int tid = threadIdx.x;

    for (int i = tid; i < NUM_ZONES; i += THREADS) lmax[i] = 0ull;
    __syncthreads();

    const int b    = blockIdx.y;
    const int base = blockIdx.x * PIX_PER_WG;        // multiple of 4
    const float4* __restrict__ xb = (const float4*)(x     + (size_t)b * NPIX) + (base >> 2);
    const int4*   __restrict__ zb = (const int4*)  (zones + (size_t)b * NPIX) + (base >> 2);

    for (int i = 0; i < ITERS; ++i) {
        const int vi = i * THREADS + tid;            // vec4 index within this WG's slice
        if (i + 4 < ITERS) {  // stream ahead (global_prefetch_b8)
            __builtin_prefetch(&xb[vi + 4 * THREADS], 0, 0);
            __builtin_prefetch(&zb[vi + 4 * THREADS], 0, 0);
        }
        const int4   z4 = zb[vi];                    // global_load_b128
        const float4 v4 = xb[vi];                    // global_load_b128
        const unsigned p0 = (unsigned)(base + vi * 4);
        atomicMax(&lmax[(unsigned)z4.x], pack_vi(v4.x, p0 + 0));  // ds_max_u64
        atomicMax(&lmax[(unsigned)z4.y], pack_vi(v4.y, p0 + 1));
        atomicMax(&lmax[(unsigned)z4.z], pack_vi(v4.z, p0 + 2));
        atomicMax(&lmax[(unsigned)z4.w], pack_vi(v4.w, p0 + 3));
    }
    __syncthreads();

    // Flush non-empty entries; cuts global atomics ~4x vs per-pixel scatter.
    unsigned long long* __restrict__ gz = zmax + (size_t)b * NUM_ZONES;
    for (int i = tid; i < NUM_ZONES; i += THREADS) {
        const unsigned long long m = lmax[i];
        if (m) atomicMax(&gz[i], m);                 // global_atomic_max_u64
    }
}

__global__ __launch_bounds__(THREADS) void centermask_kernel(
        const float* __restrict__ x,
        const int*   __restrict__ zones,
        const unsigned long long* __restrict__ zmax,
        float*    __restrict__ outmask,
        unsigned* __restrict__ counter) {
    const unsigned vid  = blockIdx.x * THREADS + threadIdx.x;  // vec4 id, 0 .. 4M-1
    const unsigned gid  = vid * 4u;                            // pixel id, 0 .. 16M-1
    const unsigned b    = gid >> 20;
    const unsigned p    = gid & (NPIX - 1);

    const int4   z4 = ((const int4*)  zones)[vid];             // global_load_b128
    const float4 v4 = ((const float4*)x)[vid];                 // global_load_b128
    const unsigned long long* __restrict__ zb = zmax + (size_t)b * NUM_ZONES;

    const bool c0 = (zb[(unsigned)z4.x] == pack_vi(v4.x, p + 0));
    const bool c1 = (zb[(unsigned)z4.y] == pack_vi(v4.y, p + 1));
    const bool c2 = (zb[(unsigned)z4.z] == pack_vi(v4.z, p + 2));
    const bool c3 = (zb[(unsigned)z4.w] == pack_vi(v4.w, p + 3));

    float4 m4;
    m4.x = c0 ? 1.0f : 0.0f;
    m4.y = c1 ? 1.0f : 0.0f;
    m4.z = c2 ? 1.0f : 0.0f;
    m4.w = c3 ? 1.0f : 0.0f;
    ((float4*)outmask)[vid] = m4;                              // global_store_b128

#if __has_builtin(__builtin_amdgcn_ballot_w32)
    const unsigned cnt = (unsigned)__popc(__builtin_amdgcn_ballot_w32(c0))
                       + (unsigned)__popc(__builtin_amdgcn_ballot_w32(c1))
                       + (unsigned)__popc(__builtin_amdgcn_ballot_w32(c2))
                       + (unsigned)__popc(__builtin_amdgcn_ballot_w32(c3));
#else
    const unsigned cnt = (unsigned)__popcll(__ballot(c0))
                       + (unsigned)__popcll(__ballot(c1))
                       + (unsigned)__popcll(__ballot(c2))
                       + (unsigned)__popcll(__ballot(c3));
#endif
    if ((threadIdx.x & 31u) == 0u && cnt)
        atomicAdd(counter, cnt);
}

__global__ void write_count_kernel(const unsigned* __restrict__ counter,
                                   float* __restrict__ out) {
    out[0] = (float)(*counter);   // <= 131072 centers, exact in fp32
}

extern "C" void kernel_launch(void* const* d_in, const int* in_sizes, int n_in,
                              void* d_out, int out_size, void* d_ws, size_t ws_size,
                              hipStream_t stream) {
    const float* x     = (const float*)d_in[0];   // [16,1,1024,1024] f32
    const int*   zones = (const int*)  d_in[1];   // [16,1024,1024] int (values < 8192)
    float*       out   = (float*)d_out;           // [16*2^20] mask + 1 count

    // Workspace: 16*8192 u64 zone table (1 MB) + one u32 counter.
    unsigned long long* zmax = (unsigned long long*)d_ws;
    unsigned* counter = (unsigned*)((char*)d_ws +
                        (size_t)NBATCH * NUM_ZONES * sizeof(unsigned long long));

    init_ws_kernel<<<(NBATCH * NUM_ZONES / 4 + 255) / 256, 256, 0, stream>>>(zmax, counter);

    dim3 gridA(WGS_PER_IMG, NBATCH);
    zone_max_kernel<<<gridA, THREADS, 0, stream>>>(x, zones, zmax);

    centermask_kernel<<<(NBATCH * (size_t)NPIX) / (THREADS * VEC), THREADS, 0, stream>>>(
        x, zones, zmax, out, counter);

    write_count_kernel<<<1, 1, 0, stream>>>(counter, out + (size_t)NBATCH * NPIX);
}